// NeuralMemoryCell_28200755265712
// MI455X (gfx1250) — compile-verified
//
#include <hip/hip_runtime.h>
#include <hip/hip_bf16.h>
#include <math.h>

// Problem dims (compile-time constants from the reference)
#define B_  4
#define L_  512
#define D_  512
#define H_  8
#define DH_ 64

typedef float v2f __attribute__((ext_vector_type(2)));
typedef float v8f __attribute__((ext_vector_type(8)));

// ---------------------------------------------------------------------------
// Kernel 1: projection GEMMs  O = X @ W   (X:[2048,512], W:[512,512] fp32)
// Each wave32 computes a 64(M) x 16(N) strip with V_WMMA_F32_16X16X4_F32:
// one strided B fragment is reused across 4 WMMAs (4 M-subtiles), cutting
// the expensive row-strided W loads 4x. A loads are contiguous b64.
// 32 M-strips * 32 N-tiles = 1024 wave-tiles; 8 waves/block -> grid.x = 128.
// blockIdx.z in {0,1,2} selects {Wq,Wk,Wv} -> {q,k,v}.
// ---------------------------------------------------------------------------
__global__ __launch_bounds__(256)
void proj_gemm_wmma(const float* __restrict__ X,
                    const float* __restrict__ W0,
                    const float* __restrict__ W1,
                    const float* __restrict__ W2,
                    float* __restrict__ O0,
                    float* __restrict__ O1,
                    float* __restrict__ O2)
{
    const float* W = (blockIdx.z == 0) ? W0 : (blockIdx.z == 1) ? W1 : W2;
    float*       O = (blockIdx.z == 0) ? O0 : (blockIdx.z == 1) ? O1 : O2;

    const int lane = threadIdx.x & 31;
    const int wave = threadIdx.x >> 5;
    const int tile = blockIdx.x * 8 + wave;   // 0..1023
    const int tm   = tile >> 5;               // 32 M-strips (64 rows each)
    const int tn   = tile & 31;               // 32 N-tiles
    const int m0   = tm * 64;
    const int n0   = tn * 16;

    v8f c0 = {}, c1 = {}, c2 = {}, c3 = {};

    // A (16x4 f32): lanes 0-15 -> K=0,1 ; lanes 16-31 -> K=2,3 ; M = lane&15
    // B (4x16 f32): mirrored   ; N = lane&15
    const int mrow = lane & 15;
    const int n    = n0 + (lane & 15);
    const int kb   = (lane >> 4) * 2;
    const float* A0 = X + (size_t)(m0 + mrow) * D_ + kb;
    const float* Bc = W + (size_t)kb * D_ + n;

    #pragma unroll 2
    for (int k0 = 0; k0 < D_; k0 += 4) {
        v2f b;
        b.x = Bc[(size_t)k0 * D_];
        b.y = Bc[(size_t)(k0 + 1) * D_];
        v2f a0, a1, a2, a3;
        a0.x = A0[k0];                a0.y = A0[k0 + 1];
        a1.x = A0[k0 + 16 * D_];      a1.y = A0[k0 + 1 + 16 * D_];
        a2.x = A0[k0 + 32 * D_];      a2.y = A0[k0 + 1 + 32 * D_];
        a3.x = A0[k0 + 48 * D_];      a3.y = A0[k0 + 1 + 48 * D_];
        c0 = __builtin_amdgcn_wmma_f32_16x16x4_f32(false, a0, false, b, (short)0, c0, false, false);
        c1 = __builtin_amdgcn_wmma_f32_16x16x4_f32(false, a1, false, b, (short)0, c1, false, false);
        c2 = __builtin_amdgcn_wmma_f32_16x16x4_f32(false, a2, false, b, (short)0, c2, false, false);
        c3 = __builtin_amdgcn_wmma_f32_16x16x4_f32(false, a3, false, b, (short)0, c3, false, false);
    }

    // C/D layout: VGPR v -> row sub_m0 + v + (lane/16)*8, col n0 + lane&15
    const int col   = n0 + (lane & 15);
    const int rhalf = (lane >> 4) * 8;
    #pragma unroll
    for (int v = 0; v < 8; ++v) {
        O[(size_t)(m0 +      rhalf + v) * D_ + col] = c0[v];
        O[(size_t)(m0 + 16 + rhalf + v) * D_ + col] = c1[v];
        O[(size_t)(m0 + 32 + rhalf + v) * D_ + col] = c2[v];
        O[(size_t)(m0 + 48 + rhalf + v) * D_ + col] = c3[v];
    }
}

// ---------------------------------------------------------------------------
// Kernel 2: params = X @ Wp + bp   ([2048,512]x[512,24], tiny)
// ---------------------------------------------------------------------------
__global__ __launch_bounds__(256)
void params_gemm(const float* __restrict__ X,
                 const float* __restrict__ Wp,
                 const float* __restrict__ bp,
                 float* __restrict__ P)
{
    const int t = blockIdx.x * blockDim.x + threadIdx.x;
    if (t >= (B_ * L_) * 3 * H_) return;
    const int i = t / (3 * H_);
    const int j = t % (3 * H_);
    const float* xr = X + (size_t)i * D_;
    float acc = bp[j];
    for (int k = 0; k < D_; ++k)
        acc = fmaf(xr[k], Wp[(size_t)k * (3 * H_) + j], acc);
    P[t] = acc;
}

// ---------------------------------------------------------------------------
// Kernel 3: sequential scan. Rows of (state, momentum) evolve independently:
//   vhat[d] = state[d,:]·kn ; grad[d,:] = (vhat[d]-v[d])*kn[:]
// One wave32 owns row (b,h,d), 2 f32 elems/lane in registers; 2048 rows,
// 8 waves/block -> 256 blocks. Reductions are shfl_xor butterflies.
// The 536MB state/momentum streams use NON-TEMPORAL b64 stores so they do
// not thrash the 192MB L2 that is caching the shared q/k/v reads.
// ---------------------------------------------------------------------------
__global__ __launch_bounds__(256)
void scan_kernel(const float* __restrict__ qb,
                 const float* __restrict__ kb,
                 const float* __restrict__ vb,
                 const float* __restrict__ pb,
                 const float* __restrict__ s_init,
                 const float* __restrict__ m_init,
                 float* __restrict__ yout,   // [B,L,H*Dh]
                 float* __restrict__ sout,   // [B,L,H,Dh,Dh]
                 float* __restrict__ mout)   // [B,L,H,Dh,Dh]
{
    const int lane = threadIdx.x & 31;
    const int wave = threadIdx.x >> 5;
    const int r    = blockIdx.x * 8 + wave;  // row id = ((b*H)+h)*Dh + d
    const int d    = r & (DH_ - 1);
    const int bh   = r >> 6;
    const int h    = bh & (H_ - 1);
    const int b    = bh >> 3;
    const int e0   = lane * 2;               // this lane's 2 columns

    float2 s2 = *reinterpret_cast<const float2*>(s_init + (size_t)r * DH_ + e0);
    float2 m2 = *reinterpret_cast<const float2*>(m_init + (size_t)r * DH_ + e0);
    float s0 = s2.x, s1 = s2.y, mm0 = m2.x, mm1 = m2.y;

    for (int t = 0; t < L_; ++t) {
        const size_t rowoff = ((size_t)b * L_ + t) * (H_ * DH_) + (size_t)h * DH_;
        const float2 q2 = *reinterpret_cast<const float2*>(qb + rowoff + e0);
        const float2 k2 = *reinterpret_cast<const float2*>(kb + rowoff + e0);
        const float  vd = vb[rowoff + d];

        const size_t poff = ((size_t)b * L_ + t) * (3 * H_) + h;
        const float pa = pb[poff];
        const float pe = pb[poff + H_];
        const float pt = pb[poff + 2 * H_];
        const float alpha = 1.0f / (1.0f + expf(-pa));
        const float eta   = 1.0f / (1.0f + expf(-pe));
        const float theta = (pt > 20.0f) ? pt : log1pf(expf(pt));

        // three simultaneous 64-wide reductions (||k||^2, state·k, state·q)
        float sq = k2.x * k2.x + k2.y * k2.y;
        float pk = s0 * k2.x + s1 * k2.y;
        float pq = s0 * q2.x + s1 * q2.y;
        #pragma unroll
        for (int off = 16; off > 0; off >>= 1) {
            sq += __shfl_xor(sq, off, 32);
            pk += __shfl_xor(pk, off, 32);
            pq += __shfl_xor(pq, off, 32);
        }

        const float inv  = 1.0f / fmaxf(sqrtf(sq), 1e-12f);
        const float vhat = pk * inv;              // state · (k/||k||)
        const float diff = vhat - vd;
        const float g0   = diff * k2.x * inv;
        const float g1   = diff * k2.y * inv;

        mm0 = eta * mm0 - theta * g0;
        mm1 = eta * mm1 - theta * g1;
        const float oma = 1.0f - alpha;
        s0 = oma * s0 + mm0;
        s1 = oma * s1 + mm1;

        if (lane == 0)                            // y uses pre-update state
            yout[rowoff + d] = pq;

        const size_t so = (((size_t)(b * L_ + t) * H_ + h) * DH_ + d) * DH_ + e0;
        v2f sv; sv.x = s0;  sv.y = s1;
        v2f mv; mv.x = mm0; mv.y = mm1;
        __builtin_nontemporal_store(sv, reinterpret_cast<v2f*>(sout + so));
        __builtin_nontemporal_store(mv, reinterpret_cast<v2f*>(mout + so));
    }
}

// ---------------------------------------------------------------------------
extern "C" void kernel_launch(void* const* d_in, const int* in_sizes, int n_in,
                              void* d_out, int out_size, void* d_ws, size_t ws_size,
                              hipStream_t stream)
{
    (void)in_sizes; (void)n_in; (void)out_size; (void)ws_size;

    const float* X   = (const float*)d_in[0];  // inputs [B,L,D]
    const float* S0  = (const float*)d_in[1];  // prev_state [B,H,Dh,Dh]
    const float* M0  = (const float*)d_in[2];  // prev_momentum
    const float* Wq  = (const float*)d_in[3];
    const float* Wk  = (const float*)d_in[4];
    const float* Wv  = (const float*)d_in[5];
    const float* Wp  = (const float*)d_in[6];
    const float* bp  = (const float*)d_in[7];

    float* out  = (float*)d_out;
    const size_t n_read  = (size_t)B_ * L_ * H_ * DH_;          // 1,048,576
    const size_t n_state = (size_t)B_ * L_ * H_ * DH_ * DH_;    // 67,108,864
    float* yout = out;
    float* sout = out + n_read;
    float* mout = sout + n_state;

    float* qb   = (float*)d_ws;               // 4 MB
    float* kbuf = qb + n_read;                // 4 MB
    float* vbuf = kbuf + n_read;              // 4 MB
    float* pbuf = vbuf + n_read;              // 192 KB

    // q/k/v projections via f32 WMMA (64x16 per wave, B-fragment reuse)
    dim3 gproj(128, 1, 3);
    proj_gemm_wmma<<<gproj, 256, 0, stream>>>(X, Wq, Wk, Wv, qb, kbuf, vbuf);

    // gate params
    const int np = (B_ * L_) * 3 * H_;
    params_gemm<<<(np + 255) / 256, 256, 0, stream>>>(X, Wp, bp, pbuf);

    // row-parallel scan: 2048 rows, 8 waves/block
    scan_kernel<<<(B_ * H_ * DH_) / 8, 256, 0, stream>>>(
        qb, kbuf, vbuf, pbuf, S0, M0, yout, sout, mout);
}